// EEGnaive_82240033784298
// MI455X (gfx1250) — compile-verified
//
#include <hip/hip_runtime.h>
#include <hip/hip_bf16.h>

typedef float v2f __attribute__((ext_vector_type(2)));
typedef float v8f __attribute__((ext_vector_type(8)));

#define BSZ   32768
#define EPS   1e-5f

// ---------------------------------------------------------------------------
// WMMA f32 GEMM: Y[M,N] = tanh(X[M,K] * W[K,N] + bias[N]).
// One wave per 64x16 output macro-tile (4 accumulators share each B fragment,
// issued back-to-back with no inter-WMMA dependency). K stepped by 4 with
// V_WMMA_F32_16X16X4_F32 (full fp32 precision on the matrix pipe).
// Edge columns handled by clamped address + 0/1 mask (no exec divergence in
// the hot loop -> WMMA's EXEC==all-ones requirement holds trivially).
// grid = (M/64, ceil(N/16)), block = 32.
// ---------------------------------------------------------------------------
#define MT 4
__global__ void wmma_gemm_bias_tanh(const float* __restrict__ X,
                                    const float* __restrict__ W,
                                    const float* __restrict__ bias,
                                    float* __restrict__ Y,
                                    int K, int N) {
    const int lane = threadIdx.x;          // 0..31
    const int half = lane >> 4;            // 0 or 1
    const int l15  = lane & 15;
    const int m0   = blockIdx.x * (16 * MT);
    const int n0   = blockIdx.y * 16;

    const int  bcol   = n0 + l15;                   // B/D column for this lane
    const int  bcolc  = (bcol < N) ? bcol : (N - 1); // clamped (always valid)
    const float bmask = (bcol < N) ? 1.0f : 0.0f;

    v8f acc[MT];
#pragma unroll
    for (int t = 0; t < MT; ++t) acc[t] = (v8f){};

    for (int k0 = 0; k0 < K; k0 += 4) {
        const int ks = k0 + half * 2;   // this lane's K pair (A and B layouts match)
        // B 4x16 fragment: VGPR0 -> K=ks, VGPR1 -> K=ks+1 (column bcol)
        v2f b;
        b.x = W[(ks + 0) * N + bcolc] * bmask;
        b.y = W[(ks + 1) * N + bcolc] * bmask;
#pragma unroll
        for (int t = 0; t < MT; ++t) {
            // A 16x4 fragment, aligned 8B load (K even, ks even)
            v2f a = *(const v2f*)(X + (m0 + t * 16 + l15) * K + ks);
            acc[t] = __builtin_amdgcn_wmma_f32_16x16x4_f32(
                /*neg_a=*/false, a, /*neg_b=*/false, b,
                /*c_mod=*/(short)0, acc[t], /*reuse_a=*/false, /*reuse_b=*/false);
        }
    }

    if (bcol < N) {
        const float bb = bias[bcol];
#pragma unroll
        for (int t = 0; t < MT; ++t) {
            const int rbase = m0 + t * 16 + half * 8;   // D: hi lanes hold rows +8
#pragma unroll
            for (int v = 0; v < 8; ++v) {
                Y[(rbase + v) * N + bcol] = tanhf(acc[t][v] + bb);
            }
        }
    }
}

// ---------------------------------------------------------------------------
// filt[b][j] = tanh(sum_k H[b][k] * w2[k][j] + b2[j]),  H:[B,35], w2:[35,7]
// ---------------------------------------------------------------------------
__global__ void filt_kernel(const float* __restrict__ H,
                            const float* __restrict__ w2,
                            const float* __restrict__ b2,
                            float* __restrict__ filt) {
    int b = blockIdx.x * blockDim.x + threadIdx.x;
    if (b >= BSZ) return;
    float h[35];
    const float* hr = H + b * 35;
#pragma unroll
    for (int k = 0; k < 35; ++k) h[k] = hr[k];
#pragma unroll
    for (int j = 0; j < 7; ++j) {
        float s = b2[j];
#pragma unroll
        for (int k = 0; k < 35; ++k) s += h[k] * w2[k * 7 + j];
        filt[b * 7 + j] = tanhf(s);
    }
}

// ---------------------------------------------------------------------------
// Per-sample 7-tap cross-correlation, fused with reshape+crop:
// eeg[b][c][t] = sum_k raw[b][c*63 + t + k] * filt[b][k]   (t in 0..56)
// (padding=3 cancels against the crop; all accesses are in-bounds)
// ---------------------------------------------------------------------------
__global__ void xcorr_kernel(const float* __restrict__ raw,
                             const float* __restrict__ filt,
                             float* __restrict__ eeg) {
    int idx = blockIdx.x * blockDim.x + threadIdx.x;   // B*684 exact
    int b = idx / 684;
    int r = idx - b * 684;
    int c = r / 57;
    int t = r - c * 57;
    const float* f = filt + b * 7;
    const float* x = raw + b * 756 + c * 63 + t;
    float s = f[0]*x[0] + f[1]*x[1] + f[2]*x[2] + f[3]*x[3]
            + f[4]*x[4] + f[5]*x[5] + f[6]*x[6];
    eeg[idx] = s;
}

// ---------------------------------------------------------------------------
// Temporal conv1 (both branches): in [B,12,57] -> pre-BN out [B,16,18],
// plus per-channel sum/sumsq (batch-norm training stats, count = B*18).
// block=256: tid&15 -> out channel, blockIdx*16 + tid>>4 -> (b,ot) pair.
// ---------------------------------------------------------------------------
__global__ void conv1_t_stats(const float* __restrict__ in,
                              const float* __restrict__ w,   // [16,12,ksz]
                              const float* __restrict__ bias,
                              float* __restrict__ outp,      // [B,16,18]
                              float* __restrict__ stats,     // sum[16], sumsq[16]
                              int ksz, int dil) {
    __shared__ float wl[16 * 12 * 6];
    __shared__ float ssum[16], ssq[16];
    const int tid = threadIdx.x;
    const int nw = 16 * 12 * ksz;
    for (int i = tid; i < nw; i += 256) wl[i] = w[i];
    if (tid < 16) { ssum[tid] = 0.0f; ssq[tid] = 0.0f; }
    __syncthreads();

    const int oc = tid & 15;
    const int p  = blockIdx.x * 16 + (tid >> 4);   // 0..B*18-1 exact
    const int b  = p / 18;
    const int ot = p - b * 18;
    const float* row = in + b * 684 + ot * 3;
    const float* wp  = wl + oc * 12 * ksz;
    float s = bias[oc];
    for (int ic = 0; ic < 12; ++ic) {
        const float* r = row + ic * 57;
        for (int k = 0; k < ksz; ++k) s += r[k * dil] * wp[ic * ksz + k];
    }
    outp[b * 288 + oc * 18 + ot] = s;
    atomicAdd(&ssum[oc], s);
    atomicAdd(&ssq[oc], s * s);
    __syncthreads();
    if (tid < 16) {
        atomicAdd(&stats[tid], ssum[tid]);
        atomicAdd(&stats[16 + tid], ssq[tid]);
    }
}

// ---------------------------------------------------------------------------
// Channel mixer: eeg_c[b][t][d] = relu(sum_c eeg[b][c][t]*c0_w[c][d] + c0_b[d])
// ---------------------------------------------------------------------------
__global__ void c0_kernel(const float* __restrict__ eeg,
                          const float* __restrict__ c0w,   // [12,9]
                          const float* __restrict__ c0b,
                          float* __restrict__ eeg_c) {     // [B,57,9]
    int idx = blockIdx.x * blockDim.x + threadIdx.x;       // B*513 exact
    int b = idx / 513;
    int r = idx - b * 513;
    int t = r / 9;
    int d = r - t * 9;
    float s = c0b[d];
    const float* e = eeg + b * 684 + t;
#pragma unroll
    for (int c = 0; c < 12; ++c) s += e[c * 57] * c0w[c * 9 + d];
    eeg_c[idx] = fmaxf(s, 0.0f);
}

// ---------------------------------------------------------------------------
// Channel-branch conv1: eeg_c [B,57,9] (*) w [16,1,5,9] stride (3,1)
//   -> pre-BN [B,16,18], + stats.
// ---------------------------------------------------------------------------
__global__ void conv1_c_stats(const float* __restrict__ in,
                              const float* __restrict__ w,   // [16,45]
                              const float* __restrict__ bias,
                              float* __restrict__ outp,      // [B,16,18]
                              float* __restrict__ stats) {
    __shared__ float wl[16 * 45];
    __shared__ float ssum[16], ssq[16];
    const int tid = threadIdx.x;
    for (int i = tid; i < 720; i += 256) wl[i] = w[i];
    if (tid < 16) { ssum[tid] = 0.0f; ssq[tid] = 0.0f; }
    __syncthreads();

    const int oc = tid & 15;
    const int p  = blockIdx.x * 16 + (tid >> 4);
    const int b  = p / 18;
    const int oh = p - b * 18;
    const float* base = in + b * 513 + oh * 27;   // oh*3 rows of width 9
    const float* wp = wl + oc * 45;
    float s = bias[oc];
#pragma unroll
    for (int kh = 0; kh < 5; ++kh)
#pragma unroll
        for (int kw = 0; kw < 9; ++kw)
            s += base[kh * 9 + kw] * wp[kh * 9 + kw];
    outp[b * 288 + oc * 18 + oh] = s;
    atomicAdd(&ssum[oc], s);
    atomicAdd(&ssq[oc], s * s);
    __syncthreads();
    if (tid < 16) {
        atomicAdd(&stats[tid], ssum[tid]);
        atomicAdd(&stats[16 + tid], ssq[tid]);
    }
}

// ---------------------------------------------------------------------------
// BN finalize: 3 layers x 16 channels -> (mean, invstd).  48 threads.
// ---------------------------------------------------------------------------
__global__ void bn_finalize(const float* __restrict__ stats,
                            float* __restrict__ mi, float inv_count) {
    int t = threadIdx.x;
    if (t >= 48) return;
    int layer = t >> 4, ch = t & 15;
    float mean = stats[layer * 32 + ch] * inv_count;
    float var  = stats[layer * 32 + 16 + ch] * inv_count - mean * mean;
    mi[layer * 32 + ch]      = mean;
    mi[layer * 32 + 16 + ch] = rsqrtf(var + EPS);
}

// ---------------------------------------------------------------------------
// conv2 (all branches): BN1+ReLU applied inline to pre1 [B,16,18], conv to
// pre-BN2 [B,16,5], + stats (count = B*5).
// ---------------------------------------------------------------------------
__global__ void conv2_stats(const float* __restrict__ pre1,  // [B,16,18]
                            const float* __restrict__ mi,    // mean[16],invstd[16]
                            const float* __restrict__ g,
                            const float* __restrict__ be,
                            const float* __restrict__ w,     // [16,16,ksz]
                            const float* __restrict__ bias,
                            float* __restrict__ pre2,        // [B,16,5]
                            float* __restrict__ stats,
                            int ksz, int dil) {
    __shared__ float wl[16 * 16 * 6];
    __shared__ float bsc[16], bsh[16];
    __shared__ float ssum[16], ssq[16];
    const int tid = threadIdx.x;
    const int nw = 256 * ksz;
    for (int i = tid; i < nw; i += 256) wl[i] = w[i];
    if (tid < 16) {
        float sc = mi[16 + tid] * g[tid];          // invstd * gamma
        bsc[tid] = sc;
        bsh[tid] = be[tid] - mi[tid] * sc;         // beta - mean*scale
        ssum[tid] = 0.0f; ssq[tid] = 0.0f;
    }
    __syncthreads();

    const int oc = tid & 15;
    const int p  = blockIdx.x * 16 + (tid >> 4);   // 0..B*5-1 exact
    const int b  = p / 5;
    const int ot = p - b * 5;
    const float* in = pre1 + b * 288 + ot * 3;
    const float* wp = wl + oc * 16 * ksz;
    float s = bias[oc];
    for (int ic = 0; ic < 16; ++ic) {
        const float sc = bsc[ic], sh = bsh[ic];
        const float* r = in + ic * 18;
        for (int k = 0; k < ksz; ++k) {
            float a = fmaxf(r[k * dil] * sc + sh, 0.0f);
            s += a * wp[ic * ksz + k];
        }
    }
    pre2[b * 80 + oc * 5 + ot] = s;
    atomicAdd(&ssum[oc], s);
    atomicAdd(&ssq[oc], s * s);
    __syncthreads();
    if (tid < 16) {
        atomicAdd(&stats[tid], ssum[tid]);
        atomicAdd(&stats[16 + tid], ssq[tid]);
    }
}

// ---------------------------------------------------------------------------
// BN2 + ReLU, scatter into feats [B,320] (c1 branch writes twice: source bug).
// ---------------------------------------------------------------------------
__global__ void apply2(const float* __restrict__ pre2,   // [B,16,5]
                       const float* __restrict__ mi,
                       const float* __restrict__ g,
                       const float* __restrict__ be,
                       float* __restrict__ feats,
                       int off1, int off2) {
    int idx = blockIdx.x * blockDim.x + threadIdx.x;     // B*80 exact
    int b = idx / 80;
    int r = idx - b * 80;
    int oc = r / 5;
    float sc = mi[16 + oc] * g[oc];
    float sh = be[oc] - mi[oc] * sc;
    float v = fmaxf(pre2[idx] * sc + sh, 0.0f);
    feats[b * 320 + off1 + r] = v;
    if (off2 >= 0) feats[b * 320 + off2 + r] = v;
}

__global__ void zero_stats(float* __restrict__ s) {
    int t = threadIdx.x;
    if (t < 512) s[t] = 0.0f;
}

// ---------------------------------------------------------------------------
extern "C" void kernel_launch(void* const* d_in, const int* in_sizes, int n_in,
                              void* d_out, int out_size, void* d_ws, size_t ws_size,
                              hipStream_t stream) {
    (void)in_sizes; (void)n_in; (void)out_size; (void)ws_size;
    const float* raw   = (const float*)d_in[0];
    const float* eegf  = (const float*)d_in[1];
    const float* fg_w1 = (const float*)d_in[2];
    const float* fg_b1 = (const float*)d_in[3];
    const float* fg_w2 = (const float*)d_in[4];
    const float* fg_b2 = (const float*)d_in[5];
    const float* t1_w1 = (const float*)d_in[6];
    const float* t1_b1 = (const float*)d_in[7];
    const float* t1_g1 = (const float*)d_in[8];
    const float* t1_be1= (const float*)d_in[9];
    const float* t1_w2 = (const float*)d_in[10];
    const float* t1_b2 = (const float*)d_in[11];
    const float* t1_g2 = (const float*)d_in[12];
    const float* t1_be2= (const float*)d_in[13];
    const float* t2_w1 = (const float*)d_in[14];
    const float* t2_b1 = (const float*)d_in[15];
    const float* t2_g1 = (const float*)d_in[16];
    const float* t2_be1= (const float*)d_in[17];
    const float* t2_w2 = (const float*)d_in[18];
    const float* t2_b2 = (const float*)d_in[19];
    const float* t2_g2 = (const float*)d_in[20];
    const float* t2_be2= (const float*)d_in[21];
    const float* c0_w  = (const float*)d_in[22];
    const float* c0_b  = (const float*)d_in[23];
    const float* c1_w1 = (const float*)d_in[24];
    const float* c1_b1 = (const float*)d_in[25];
    const float* c1_g1 = (const float*)d_in[26];
    const float* c1_be1= (const float*)d_in[27];
    const float* c1_w2 = (const float*)d_in[28];
    const float* c1_b2 = (const float*)d_in[29];
    const float* c1_g2 = (const float*)d_in[30];
    const float* c1_be2= (const float*)d_in[31];
    const float* fc_w  = (const float*)d_in[32];
    const float* fc_b  = (const float*)d_in[33];
    float* out = (float*)d_out;

    // workspace layout (floats)
    float* ws = (float*)d_ws;
    float* H       = ws;                     // B*35
    float* filt    = H       + BSZ * 35;     // B*7
    float* eeg     = filt    + BSZ * 7;      // B*684
    float* eeg_c   = eeg     + BSZ * 684;    // B*513
    float* pre1_t1 = eeg_c   + BSZ * 513;    // B*288
    float* pre1_t2 = pre1_t1 + BSZ * 288;
    float* pre1_c1 = pre1_t2 + BSZ * 288;
    float* pre2_t1 = pre1_c1 + BSZ * 288;    // B*80
    float* pre2_t2 = pre2_t1 + BSZ * 80;
    float* pre2_c1 = pre2_t2 + BSZ * 80;
    float* feats   = pre2_c1 + BSZ * 80;     // B*320
    float* stats   = feats   + BSZ * 320;    // 512 floats
    float* statsA  = stats;                  // 3 x (sum16, sq16)
    float* statsB  = stats + 96;
    float* miA     = stats + 192;            // 3 x (mean16, invstd16)
    float* miB     = stats + 288;

    zero_stats<<<1, 512, 0, stream>>>(stats);

    // 1) filter generator, layer 1 (WMMA f32): H = tanh(eegf @ fg_w1 + fg_b1)
    wmma_gemm_bias_tanh<<<dim3(BSZ / 64, 3), 32, 0, stream>>>(eegf, fg_w1, fg_b1, H, 168, 35);
    // layer 2 (tiny): filt = tanh(H @ fg_w2 + fg_b2)
    filt_kernel<<<BSZ / 256, 256, 0, stream>>>(H, fg_w2, fg_b2, filt);

    // 2) per-sample cross-correlation + reshape + crop -> eeg [B,12,57]
    xcorr_kernel<<<(BSZ * 684) / 256, 256, 0, stream>>>(raw, filt, eeg);

    // 3) conv1 of each branch (+ BN stats)
    conv1_t_stats<<<(BSZ * 18) / 16, 256, 0, stream>>>(eeg, t1_w1, t1_b1, pre1_t1, statsA + 0, 6, 1);
    conv1_t_stats<<<(BSZ * 18) / 16, 256, 0, stream>>>(eeg, t2_w1, t2_b1, pre1_t2, statsA + 32, 3, 2);
    c0_kernel<<<(BSZ * 513) / 256, 256, 0, stream>>>(eeg, c0_w, c0_b, eeg_c);
    conv1_c_stats<<<(BSZ * 18) / 16, 256, 0, stream>>>(eeg_c, c1_w1, c1_b1, pre1_c1, statsA + 64);

    bn_finalize<<<1, 64, 0, stream>>>(statsA, miA, 1.0f / (float)(BSZ * 18));

    // 4) conv2 of each branch (BN1+ReLU fused, + BN2 stats)
    conv2_stats<<<(BSZ * 5) / 16, 256, 0, stream>>>(pre1_t1, miA + 0,  t1_g1, t1_be1, t1_w2, t1_b2, pre2_t1, statsB + 0, 6, 1);
    conv2_stats<<<(BSZ * 5) / 16, 256, 0, stream>>>(pre1_t2, miA + 32, t2_g1, t2_be1, t2_w2, t2_b2, pre2_t2, statsB + 32, 3, 2);
    conv2_stats<<<(BSZ * 5) / 16, 256, 0, stream>>>(pre1_c1, miA + 64, c1_g1, c1_be1, c1_w2, c1_b2, pre2_c1, statsB + 64, 5, 1);

    bn_finalize<<<1, 64, 0, stream>>>(statsB, miB, 1.0f / (float)(BSZ * 5));

    // 5) BN2+ReLU -> feats [B,320]; c1 duplicated at offsets 160 and 240
    apply2<<<(BSZ * 80) / 256, 256, 0, stream>>>(pre2_t1, miB + 0,  t1_g2, t1_be2, feats, 0,  -1);
    apply2<<<(BSZ * 80) / 256, 256, 0, stream>>>(pre2_t2, miB + 32, t2_g2, t2_be2, feats, 80, -1);
    apply2<<<(BSZ * 80) / 256, 256, 0, stream>>>(pre2_c1, miB + 64, c1_g2, c1_be2, feats, 160, 240);

    // 6) fc (WMMA f32): out = tanh(feats @ fc_w + fc_b)
    wmma_gemm_bias_tanh<<<dim3(BSZ / 64, 4), 32, 0, stream>>>(feats, fc_w, fc_b, out, 320, 50);
}